// TokenGSM_57655640981778
// MI455X (gfx1250) — compile-verified
//
#include <hip/hip_runtime.h>
#include <hip/hip_bf16.h>

// Problem constants (fixed by the reference): B=2, T=256, N=1024, C=256.
#define TDIM   256                 // T (= GEMM M and K)
#define NCOL   262144              // N*C columns per batch
#define CCH    256                 // channels C
#define SHIFT  64
#define NFRAG_F 64                 // floats per A-fragment in LDS (32 lanes * 2)
#define GROUPS  8                  // column-groups per block (W staged once, reused 8x)

typedef __attribute__((ext_vector_type(2))) float v2f;
typedef __attribute__((ext_vector_type(8))) float v8f;

// Each wave computes a 16-column tile for ALL 256 s-rows:
//   G = W(256x256) @ X(256x16 col tile), via 64 x 16 V_WMMA_F32_16X16X4_F32.
// W is staged into LDS in WMMA A-fragment order so the 16 per-k-step fragment
// reads are ds_load_b64 with immediate offsets (m*256B), one base add per step.
__global__ __launch_bounds__(256) void TokenGSM_kernel(
    const float* __restrict__ x,     // [B, T, N, C]
    const float* __restrict__ Wg,    // [T, T]
    const float* __restrict__ bias,  // [T]
    float* __restrict__ out)         // [B, T, N, C]
{
    extern __shared__ float lds[];               // 65536 W-frag floats + 256 bias
    float* ldsW = lds;                           // [(kt*16+m)*64 + lane*2 + p]
    float* ldsB = lds + 1024 * NFRAG_F;

    const int tid    = threadIdx.x;
    const int wave   = tid >> 5;
    const int lane   = tid & 31;
    const int laneHi = lane >> 4;                // 0: lanes 0-15, 1: lanes 16-31
    const int laneLo = lane & 15;

    // ---- Stage W in A-fragment layout: frag f = kt*16 + m; lane l holds
    //      W[16*m + (l&15)][4*kt + 2*(l>>4) + {0,1}] in 2 consecutive dwords ----
    for (int f = wave; f < 1024; f += 8) {
        const int m  = f & 15;
        const int kt = f >> 4;
        const int s  = 16 * m + laneLo;
        const int t0 = 4 * kt + 2 * laneHi;      // even -> 8B-aligned global read
        v2f w = *(const v2f*)(Wg + s * TDIM + t0);
        *(v2f*)(ldsW + f * NFRAG_F + lane * 2) = w;
    }
    ldsB[tid] = bias[tid];                       // blockDim == 256 == T
    __syncthreads();

    const float* afrag_lane = ldsW + lane * 2;   // per-lane fragment base

    for (int grp = 0; grp < GROUPS; ++grp) {
        // Global column tile for this wave (16 columns), across both batches.
        const long jtile = ((long)blockIdx.x * GROUPS + grp) * 8 + wave;
        const long jg    = jtile * 16;
        const int  b     = (int)(jg >> 18);      // / NCOL
        const int  j     = (int)(jg & (NCOL - 1));
        const float* xb   = x   + (long)b * TDIM * NCOL;
        float*       outb = out + (long)b * TDIM * NCOL;

        const int    col  = j + laneLo;          // this lane's column
        const float* xcol = xb + col;            // element (t=0, col)

        v8f acc[16];
        const v8f vzero = {0.f, 0.f, 0.f, 0.f, 0.f, 0.f, 0.f, 0.f};
#pragma unroll
        for (int m = 0; m < 16; ++m) acc[m] = vzero;

        // ---- K loop: 64 steps of k=4; one B-frag load feeds 16 WMMAs ----
        for (int kt = 0; kt < 64; ++kt) {
            const int t0 = kt * 4 + 2 * laneHi;  // B-frag rows: .x=K(t0), .y=K(t0+1)
            v2f bfrag;
            bfrag.x = xcol[(long)t0 * NCOL];
            bfrag.y = xcol[(long)(t0 + 1) * NCOL];

            const float* ab = afrag_lane + kt * (16 * NFRAG_F);  // += 4096B per step
#pragma unroll
            for (int m = 0; m < 16; ++m) {
                v2f afrag = *(const v2f*)(ab + m * NFRAG_F);     // imm offset m*256B
                acc[m] = __builtin_amdgcn_wmma_f32_16x16x4_f32(
                    /*neg_a=*/false, afrag, /*neg_b=*/false, bfrag,
                    /*c_mod=*/(short)0, acc[m], /*reuse_a=*/false, /*reuse_b=*/false);
            }
        }

        // ---- Epilogue: g = sigmoid(acc + bias[s]); out = xs + g*(xsh - xs) ----
        // 16-aligned column tile => uniform shift class per wave (64 % 16 == 0).
        const int c0  = col & (CCH - 1);
        const int cls = (c0 < SHIFT) ? 0 : ((c0 < 2 * SHIFT) ? 1 : 2);

#pragma unroll
        for (int m = 0; m < 16; ++m) {
#pragma unroll
            for (int r = 0; r < 8; ++r) {
                const int s = 16 * m + r + 8 * laneHi;  // C/D: M = r + 8*(lane>=16)
                const float v = acc[m][r] + ldsB[s];
                const float g = 1.0f / (1.0f + __expf(-v));
                const float xs = xcol[(long)s * NCOL];
                float xsh;
                if (cls == 0)      xsh = (s == 0)        ? 0.0f : xcol[(long)(s - 1) * NCOL];
                else if (cls == 1) xsh = (s == TDIM - 1) ? 0.0f : xcol[(long)(s + 1) * NCOL];
                else               xsh = xs;
                outb[(long)s * NCOL + col] = fmaf(g, xsh - xs, xs);
            }
        }
    }
}

extern "C" void kernel_launch(void* const* d_in, const int* in_sizes, int n_in,
                              void* d_out, int out_size, void* d_ws, size_t ws_size,
                              hipStream_t stream) {
    const float* x    = (const float*)d_in[0];   // [2,256,1024,256] f32
    const float* W    = (const float*)d_in[1];   // [256,256] f32
    const float* bias = (const float*)d_in[2];   // [256] f32
    float*       out  = (float*)d_out;           // [2,256,1024,256] f32

    // 2 batches * (262144/16) tiles = 32768 wave-tiles = 512 blocks * 8 waves * 8 groups.
    const int nBlocks = (2 * NCOL / 16) / (8 * GROUPS);                  // 512
    const size_t ldsBytes = (size_t)(1024 * NFRAG_F + TDIM) * sizeof(float); // 257 KB

    TokenGSM_kernel<<<dim3(nBlocks), dim3(256), ldsBytes, stream>>>(x, W, bias, out);
}